// TopTPercentChannelGate_22866405883929
// MI455X (gfx1250) — compile-verified
//
#include <hip/hip_runtime.h>
#include <math.h>

// TopTPercentChannelGate for MI455X (gfx1250, wave32).
// Pass 1: per-(b,c) exact top-k mean/max via in-LDS 11/11/10-bit radix select,
//         row fetched with global_load_async_to_lds_b128 (ASYNCcnt DMA).
// Pass 2: single-workgroup fp32 WMMA MLP (V_WMMA_F32_16X16X4_F32) -> sigmoid gate.
// Pass 3: streaming x * gate.

typedef __attribute__((ext_vector_type(2))) float v2f;
typedef __attribute__((ext_vector_type(8))) float v8f;
typedef __attribute__((ext_vector_type(4))) int   v4i;
typedef __attribute__((address_space(1))) v4i*    gv4i_p;  // global
typedef __attribute__((address_space(3))) v4i*    lv4i_p;  // LDS

#define B_    32
#define C_    128
#define HID_  8
#define HW_   16384          // 128*128
#define TOPK_ 1638           // round(16384 * 0.1)
#define NBIN_ 2048           // 11-bit histogram bins

// Monotonic float<->uint key mapping (descending float == descending key).
__device__ __forceinline__ unsigned f2key(float f) {
  unsigned u = __float_as_uint(f);
  return (u & 0x80000000u) ? ~u : (u | 0x80000000u);
}
__device__ __forceinline__ float key2f(unsigned k) {
  unsigned u = (k & 0x80000000u) ? (k ^ 0x80000000u) : ~k;
  return __uint_as_float(u);
}

// Parallel top-down digit select over a 2048-bin histogram.
// In:  hist[0..2047], st[0] = kk (rank from the top, >= 1).
// Out: st[0] = remaining rank inside winning bin, st[1] = winning digit.
// chunk[] is 256-entry scratch. 256 threads.
__device__ __forceinline__ void select_digit(unsigned* __restrict__ hist,
                                             unsigned* __restrict__ chunk,
                                             unsigned* __restrict__ st,
                                             int tid) {
  unsigned s = 0;
#pragma unroll
  for (int b = 0; b < 8; ++b) s += hist[tid * 8 + b];
  chunk[tid] = s;
  __syncthreads();
  // Inclusive suffix scan: chunk[c] = sum_{c' >= c} chunkSum[c']
  for (int off = 1; off < 256; off <<= 1) {
    unsigned v   = chunk[tid];
    unsigned add = (tid + off < 256) ? chunk[tid + off] : 0u;
    __syncthreads();
    chunk[tid] = v + add;
    __syncthreads();
  }
  const unsigned kk        = st[0];
  const unsigned suff      = chunk[tid];
  const unsigned suff_next = (tid < 255) ? chunk[tid + 1] : 0u;
  if (suff >= kk && suff_next < kk) {        // exactly one winner
    unsigned run = suff_next;
    for (int b = 7; b >= 0; --b) {
      unsigned h = hist[tid * 8 + b];
      if (run + h >= kk) { st[0] = kk - run; st[1] = (unsigned)(tid * 8 + b); break; }
      run += h;
    }
  }
  __syncthreads();
}

// ---------------------------------------------------------------------------
// Kernel 1: one workgroup (256 threads = 8 wave32) per (b,c) row.
// Async-DMA the 16384-value row into LDS, then:
//   sweep A: 11-bit histogram (key>>21) + row max
//   sweep B: 11-bit histogram of next digit within prefix
//   sweep C: 10-bit histogram of low digit -> exact 32-bit threshold
//   sweep D: deterministic sum of strictly-above-threshold values
// pools layout: rows 0..4095 = avg (b*C+c), rows 4096..8191 = max.
// ---------------------------------------------------------------------------
__global__ void topk_pool_kernel(const float* __restrict__ x,
                                 float* __restrict__ pools) {
  extern __shared__ unsigned smem[];
  unsigned* vals  = smem;                       // HW_ raw float bits (64 KB)
  unsigned* hist  = smem + HW_;                 // 2048 bins (8 KB)
  unsigned* chunk = smem + HW_ + NBIN_;         // 256 scan scratch (also float reduce)
  unsigned* st    = smem + HW_ + NBIN_ + 256;   // [0]=kk, [1]=digit, [2]=maxkey

  const int tid = threadIdx.x;
  const int row = blockIdx.x;                   // b*C_ + c
  const uint4* xr4 = (const uint4*)(x + (size_t)row * HW_);
  uint4* vals4 = (uint4*)vals;

  if (tid < 3) st[tid] = 0u;
  if (tid == 0) st[0] = TOPK_;
  for (int b = tid; b < NBIN_; b += 256) hist[b] = 0u;

  // Row fetch: async global->LDS DMA (bypasses VGPRs) when available.
#if __has_builtin(__builtin_amdgcn_global_load_async_to_lds_b128)
#pragma unroll
  for (int it = 0; it < HW_ / 4 / 256; ++it) {
    int j = tid + it * 256;
    __builtin_amdgcn_global_load_async_to_lds_b128(
        (gv4i_p)(xr4 + j), (lv4i_p)(vals4 + j), 0, 0);
  }
#if __has_builtin(__builtin_amdgcn_s_wait_asynccnt)
  __builtin_amdgcn_s_wait_asynccnt(0);
#else
  asm volatile("s_wait_asynccnt 0" ::: "memory");
#endif
#else
  for (int it = 0; it < HW_ / 4 / 256; ++it) {
    int j = tid + it * 256;
    vals4[j] = xr4[j];
  }
#endif
  __syncthreads();

  // Sweep A: top-11-bit histogram + max.
  unsigned lmax = 0u;
  for (int i = tid; i < HW_ / 4; i += 256) {
    uint4 r = vals4[i];
    unsigned k0 = f2key(__uint_as_float(r.x)), k1 = f2key(__uint_as_float(r.y));
    unsigned k2 = f2key(__uint_as_float(r.z)), k3 = f2key(__uint_as_float(r.w));
    atomicAdd(&hist[k0 >> 21], 1u);
    atomicAdd(&hist[k1 >> 21], 1u);
    atomicAdd(&hist[k2 >> 21], 1u);
    atomicAdd(&hist[k3 >> 21], 1u);
    unsigned m01 = k0 > k1 ? k0 : k1;
    unsigned m23 = k2 > k3 ? k2 : k3;
    unsigned m   = m01 > m23 ? m01 : m23;
    if (m > lmax) lmax = m;
  }
  atomicMax(&st[2], lmax);
  __syncthreads();

  select_digit(hist, chunk, st, tid);
  unsigned p1 = st[1];                                  // 11-bit prefix
  for (int b = tid; b < NBIN_; b += 256) hist[b] = 0u;
  __syncthreads();

  // Sweep B: middle-11-bit histogram within prefix.
  for (int i = tid; i < HW_ / 4; i += 256) {
    uint4 r = vals4[i];
    unsigned k;
    k = f2key(__uint_as_float(r.x)); if ((k >> 21) == p1) atomicAdd(&hist[(k >> 10) & 0x7FFu], 1u);
    k = f2key(__uint_as_float(r.y)); if ((k >> 21) == p1) atomicAdd(&hist[(k >> 10) & 0x7FFu], 1u);
    k = f2key(__uint_as_float(r.z)); if ((k >> 21) == p1) atomicAdd(&hist[(k >> 10) & 0x7FFu], 1u);
    k = f2key(__uint_as_float(r.w)); if ((k >> 21) == p1) atomicAdd(&hist[(k >> 10) & 0x7FFu], 1u);
  }
  __syncthreads();

  select_digit(hist, chunk, st, tid);
  unsigned p2 = (p1 << 11) | st[1];                     // 22-bit prefix
  for (int b = tid; b < NBIN_; b += 256) hist[b] = 0u;
  __syncthreads();

  // Sweep C: low-10-bit histogram within prefix -> exact threshold.
  for (int i = tid; i < HW_ / 4; i += 256) {
    uint4 r = vals4[i];
    unsigned k;
    k = f2key(__uint_as_float(r.x)); if ((k >> 10) == p2) atomicAdd(&hist[k & 0x3FFu], 1u);
    k = f2key(__uint_as_float(r.y)); if ((k >> 10) == p2) atomicAdd(&hist[k & 0x3FFu], 1u);
    k = f2key(__uint_as_float(r.z)); if ((k >> 10) == p2) atomicAdd(&hist[k & 0x3FFu], 1u);
    k = f2key(__uint_as_float(r.w)); if ((k >> 10) == p2) atomicAdd(&hist[k & 0x3FFu], 1u);
  }
  __syncthreads();

  select_digit(hist, chunk, st, tid);
  const unsigned T      = (p2 << 10) | st[1];           // exact threshold key
  const unsigned kk_rem = st[0];                        // #ties taken at T

  // Sweep D: sum of strictly-above-threshold values; deterministic tree reduce.
  float lsum = 0.f;
  for (int i = tid; i < HW_ / 4; i += 256) {
    uint4 r = vals4[i];
    unsigned k;
    k = f2key(__uint_as_float(r.x)); if (k > T) lsum += __uint_as_float(r.x);
    k = f2key(__uint_as_float(r.y)); if (k > T) lsum += __uint_as_float(r.y);
    k = f2key(__uint_as_float(r.z)); if (k > T) lsum += __uint_as_float(r.z);
    k = f2key(__uint_as_float(r.w)); if (k > T) lsum += __uint_as_float(r.w);
  }
  float* fred = (float*)chunk;
  fred[tid] = lsum;
  __syncthreads();
  for (int s = 128; s > 0; s >>= 1) {
    if (tid < s) fred[tid] += fred[tid + s];
    __syncthreads();
  }
  if (tid == 0) {
    float sum_top = fred[0] + (float)kk_rem * key2f(T);
    pools[row]           = sum_top * (1.0f / (float)TOPK_);  // avg pool
    pools[B_ * C_ + row] = key2f(st[2]);                     // max pool
  }
}

// ---------------------------------------------------------------------------
// Kernel 2: one workgroup, 128 threads (4 wave32s). Batches avg||max pools as
// a [64,128] A matrix; full-fp32 WMMA (16x16x4) MLP; sigmoid gate out.
// ---------------------------------------------------------------------------
__global__ void mlp_gate_kernel(const float* __restrict__ pools,
                                const float* __restrict__ w1,
                                const float* __restrict__ b1,
                                const float* __restrict__ w2,
                                const float* __restrict__ b2,
                                float* __restrict__ scale) {
  extern __shared__ float sm[];
  float* P  = sm;                         // [64][128] pools      (8192)
  float* W1 = sm + 8192;                  // [128][16] padded     (2048)
  float* W2 = sm + 8192 + 2048;           // [16][128] padded     (2048)
  float* Hh = sm + 8192 + 4096;           // [64][16] relu hidden (1024)
  float* O  = sm + 8192 + 4096 + 1024;    // [64][128] mlp out    (8192)

  const int tid = threadIdx.x;            // 128 threads
  for (int i = tid; i < 8192; i += 128) P[i] = pools[i];
  for (int i = tid; i < 2048; i += 128) { // w1: [128][8] -> pad N to 16
    int r = i >> 4, c = i & 15;
    W1[i] = (c < HID_) ? w1[r * HID_ + c] : 0.f;
  }
  for (int i = tid; i < 2048; i += 128) { // w2: [8][128] -> pad K to 16
    int r = i >> 7, c = i & 127;
    W2[i] = (r < HID_) ? w2[r * C_ + c] : 0.f;
  }
  __syncthreads();

  const int wave = tid >> 5;              // wave w owns rows 16w..16w+15
  const int lane = tid & 31;
  const int mn   = lane & 15;             // M (A frag) / N (B frag)
  const int koff = (lane >> 4) * 2;       // lanes 16-31 hold K+2,K+3
  const int mhi  = (lane >> 4) * 8;       // C/D: lanes 16-31 hold M=r+8

  // GEMM1: [64x128] @ [128x16] -> relu(+b1) -> Hh
  {
    v8f acc = {};
    for (int k0 = 0; k0 < 128; k0 += 4) {
      v2f a, b;
      const float* ap = P + (wave * 16 + mn) * 128 + k0 + koff;
      a[0] = ap[0];
      a[1] = ap[1];
      b[0] = W1[(k0 + koff) * 16 + mn];
      b[1] = W1[(k0 + koff + 1) * 16 + mn];
      acc = __builtin_amdgcn_wmma_f32_16x16x4_f32(false, a, false, b,
                                                  (short)0, acc, false, false);
    }
    for (int r = 0; r < 8; ++r) {
      int m = wave * 16 + mhi + r;
      float v = acc[r] + (mn < HID_ ? b1[mn] : 0.f);
      Hh[m * 16 + mn] = v > 0.f ? v : 0.f;
    }
  }
  __syncthreads();

  // GEMM2: [64x16] @ [16x128] -> O
  for (int nt = 0; nt < 8; ++nt) {
    v8f acc = {};
    for (int k0 = 0; k0 < 16; k0 += 4) {
      v2f a, b;
      const float* ap = Hh + (wave * 16 + mn) * 16 + k0 + koff;
      a[0] = ap[0];
      a[1] = ap[1];
      b[0] = W2[(k0 + koff) * 128 + nt * 16 + mn];
      b[1] = W2[(k0 + koff + 1) * 128 + nt * 16 + mn];
      acc = __builtin_amdgcn_wmma_f32_16x16x4_f32(false, a, false, b,
                                                  (short)0, acc, false, false);
    }
    for (int r = 0; r < 8; ++r) {
      int m = wave * 16 + mhi + r;
      O[m * 128 + nt * 16 + mn] = acc[r];
    }
  }
  __syncthreads();

  // gate = sigmoid(O_avg + O_max + 2*b2)   (b2 added once per mlp() call)
  for (int i = tid; i < B_ * C_; i += 128) {
    int c = i & 127;
    float att = O[i] + O[B_ * C_ + i] + 2.f * b2[c];
    scale[i] = 1.f / (1.f + expf(-att));
  }
}

// ---------------------------------------------------------------------------
// Kernel 3: streaming out = x * gate. One block per (b,c) row; gate is
// block-uniform (scalar); float4 traffic at HBM rate.
// ---------------------------------------------------------------------------
__global__ void scale_kernel(const float* __restrict__ x,
                             const float* __restrict__ scale,
                             float* __restrict__ out) {
  const int row = blockIdx.x;
  const float s = scale[row];
  const float4* xi = (const float4*)(x + (size_t)row * HW_);
  float4* oi = (float4*)(out + (size_t)row * HW_);
  for (int i = threadIdx.x; i < HW_ / 4; i += 256) {
    float4 v = xi[i];
    v.x *= s; v.y *= s; v.z *= s; v.w *= s;
    oi[i] = v;
  }
}

extern "C" void kernel_launch(void* const* d_in, const int* in_sizes, int n_in,
                              void* d_out, int out_size, void* d_ws, size_t ws_size,
                              hipStream_t stream) {
  const float* x  = (const float*)d_in[0];   // [32,128,128,128]
  const float* w1 = (const float*)d_in[1];   // [128,8]
  const float* b1 = (const float*)d_in[2];   // [8]
  const float* w2 = (const float*)d_in[3];   // [8,128]
  const float* b2 = (const float*)d_in[4];   // [128]
  float* out = (float*)d_out;

  float* pools = (float*)d_ws;               // [64][128]: avg rows 0-31, max rows 32-63
  float* gate  = pools + 2 * B_ * C_;        // [32][128]

  const size_t sh1 = (size_t)(HW_ + NBIN_ + 256 + 8) * sizeof(unsigned);  // ~73 KB
  const size_t sh2 = (size_t)(8192 + 2048 + 2048 + 1024 + 8192) * sizeof(float); // 84 KB

  topk_pool_kernel<<<B_ * C_, 256, sh1, stream>>>(x, pools);
  mlp_gate_kernel<<<1, 128, sh2, stream>>>(pools, w1, b1, w2, b2, gate);
  scale_kernel<<<B_ * C_, 256, 0, stream>>>(x, gate, out);
}